// ClusterNorm1dv5_37151467110661
// MI455X (gfx1250) — compile-verified
//
#include <hip/hip_runtime.h>

// ClusterNorm whitening for MI455X (gfx1250, wave32, WMMA + async-to-LDS).
// x:[B,D,K] f32 -> per-cluster covariance (WMMA), OAS shrinkage, C^{-1/2} via
// Newton-Schulz (all-WMMA, replaces eigh), whitening (WMMA), output [B,D,K].
// Tile staging uses GLOBAL_LOAD_ASYNC_TO_LDS_B128 (ASYNCcnt) with double
// buffering in the covariance kernel.

typedef __attribute__((ext_vector_type(2))) float v2f;
typedef __attribute__((ext_vector_type(8))) float v8f;

#define BATCH 8192
#define DDIM  64
#define KCL   64
#define LS    68          // LDS row stride (floats): 272 B -> 16B aligned rows, bank-skewed
#define NS_ITERS 20
#define BSPLIT 16         // batch split for covariance occupancy
#define BCHUNK (BATCH / BSPLIT)

// ---------------- CDNA5 async copy: global -> LDS, 16 bytes per lane --------
__device__ __forceinline__ void async_copy16(const float* g, const float* lds_p) {
  const unsigned lds_addr = (unsigned)(unsigned long long)(const void*)lds_p; // flat LDS addr low 32 = LDS offset
  const unsigned long long ga = (unsigned long long)(const void*)g;
  asm volatile("global_load_async_to_lds_b128 %0, %1, off"
               :: "v"(lds_addr), "v"(ga) : "memory");
}
__device__ __forceinline__ void wait_async0() {
  asm volatile("s_wait_asynccnt 0x0" ::: "memory");
}

__device__ __forceinline__ v8f wmma4(v2f a, v2f b, v8f c) {
  return __builtin_amdgcn_wmma_f32_16x16x4_f32(false, a, false, b, (short)0, c,
                                               false, false);
}

// Wave computes 16x16 tile (ti,tj) of A(64x64)*B(64x64), row-major LDS, stride LS.
__device__ __forceinline__ v8f lds_gemm64(const float* __restrict__ A,
                                          const float* __restrict__ Bm,
                                          int ti, int tj, int lane) {
  v8f acc = {};
  const int m  = lane & 15;
  const int kh = (lane >> 4) << 1;
#pragma unroll
  for (int c = 0; c < 16; ++c) {
    const int k0 = 4 * c + kh;
    v2f a, b;
    a.x = A[(ti * 16 + m) * LS + k0];
    a.y = A[(ti * 16 + m) * LS + k0 + 1];
    b.x = Bm[k0 * LS + tj * 16 + m];
    b.y = Bm[(k0 + 1) * LS + tj * 16 + m];
    acc = wmma4(a, b, acc);
  }
  return acc;
}

// ---------------- stage 1: column sums for the mean (two-stage, no atomics)
__global__ void __launch_bounds__(256)
mean_partial_kernel(const float* __restrict__ x, float* __restrict__ partial) {
  const int dk    = blockIdx.x * 256 + threadIdx.x;  // d*64+k
  const int chunk = blockIdx.y;                      // 0..31
  float s = 0.f;
  const int b0 = chunk * (BATCH / 32);
#pragma unroll 4
  for (int b = b0; b < b0 + BATCH / 32; ++b)
    s += x[(size_t)b * (DDIM * KCL) + dk];
  partial[(size_t)chunk * (DDIM * KCL) + dk] = s;
}

__global__ void __launch_bounds__(256)
mean_reduce_kernel(const float* __restrict__ partial, float* __restrict__ musum) {
  const int dk = blockIdx.x * 256 + threadIdx.x;
  float s = 0.f;
#pragma unroll
  for (int c = 0; c < 32; ++c) s += partial[(size_t)c * (DDIM * KCL) + dk];
  musum[dk] = s;
}

// ---------------- stage 2: center + transpose [B,D,K] -> y[K,D,B]
__global__ void __launch_bounds__(256)
center_transpose_kernel(const float* __restrict__ x, const float* __restrict__ musum,
                        float* __restrict__ y) {
  __shared__ float t[64 * LS];
  const int d = blockIdx.y, b0 = blockIdx.x * 64, tid = threadIdx.x;
  // async-stage 64x64 tile: t[i][j] = x[b0+i][d][j], 1024 x 16B segments
#pragma unroll
  for (int tseg = 0; tseg < 4; ++tseg) {
    const int s = tid + tseg * 256;
    const int i = s >> 4, q = (s & 15) << 2;
    async_copy16(x + (size_t)(b0 + i) * (DDIM * KCL) + d * KCL + q, &t[i * LS + q]);
  }
  wait_async0();
  __syncthreads();
  const float invB = 1.0f / (float)BATCH;
  for (int e = tid; e < 64 * 64; e += 256) {
    const int kk = e >> 6, i = e & 63;           // i fastest -> coalesced writes
    y[(size_t)kk * (DDIM * BATCH) + (size_t)d * BATCH + b0 + i]
        = t[i * LS + kk] - musum[d * KCL + kk] * invB;
  }
}

// ---------------- stage 3a: partial cov, double-buffered async staging
__global__ void __launch_bounds__(512)
cov_partial_kernel(const float* __restrict__ y, float* __restrict__ covp) {
  __shared__ float ys[2][64 * LS];
  const int k = blockIdx.x, bs = blockIdx.y, tid = threadIdx.x;
  const int lane = tid & 31, wave = tid >> 5;
  const int ti = wave >> 2, tj = wave & 3;
  const int m  = lane & 15;
  const int kh = (lane >> 4) << 1;
  const float* yk = y + (size_t)k * (DDIM * BATCH) + (size_t)bs * BCHUNK;

  // prologue: fill buffer 0
#pragma unroll
  for (int tseg = 0; tseg < 2; ++tseg) {
    const int s = tid + tseg * 512;
    const int dd = s >> 4, q = (s & 15) << 2;
    async_copy16(yk + (size_t)dd * BATCH + q, &ys[0][dd * LS + q]);
  }
  wait_async0();
  __syncthreads();

  v8f acc = {};
  int cur = 0;
  for (int b0 = 0; b0 < BCHUNK; b0 += 64) {
    if (b0 + 64 < BCHUNK) {                       // prefetch next tile async
#pragma unroll
      for (int tseg = 0; tseg < 2; ++tseg) {
        const int s = tid + tseg * 512;
        const int dd = s >> 4, q = (s & 15) << 2;
        async_copy16(yk + (size_t)dd * BATCH + b0 + 64 + q,
                     &ys[cur ^ 1][dd * LS + q]);
      }
    }
    const float* buf = ys[cur];
#pragma unroll
    for (int c = 0; c < 16; ++c) {
      const int k0 = 4 * c + kh;
      v2f a, b;
      a.x = buf[(ti * 16 + m) * LS + k0];         // A[d_row, b]
      a.y = buf[(ti * 16 + m) * LS + k0 + 1];
      b.x = buf[(tj * 16 + m) * LS + k0];         // B[b, e_col] = Y[e_col, b]
      b.y = buf[(tj * 16 + m) * LS + k0 + 1];
      acc = wmma4(a, b, acc);
    }
    wait_async0();
    __syncthreads();
    cur ^= 1;
  }
#pragma unroll
  for (int r = 0; r < 8; ++r) {
    const int row = ti * 16 + r + ((lane >> 4) << 3);   // C/D: M = r + 8*lane[4]
    const int col = tj * 16 + (lane & 15);
    covp[((size_t)bs * KCL + k) * 4096 + row * 64 + col] = acc[r];
  }
}

// ---------------- stage 3b: reduce partial covs, scale by 1/B
__global__ void __launch_bounds__(256)
cov_reduce_kernel(const float* __restrict__ covp, float* __restrict__ cov) {
  const int k = blockIdx.x;
  const int e = blockIdx.y * 256 + threadIdx.x;   // grid.y = 16 -> 4096 elems
  float s = 0.f;
#pragma unroll
  for (int c = 0; c < BSPLIT; ++c)
    s += covp[((size_t)c * KCL + k) * 4096 + e];
  cov[(size_t)k * 4096 + e] = s * (1.0f / (float)BATCH);
}

// ---------------- stage 4: OAS shrinkage + Newton-Schulz inverse-sqrt (all WMMA)
__global__ void __launch_bounds__(512)
ns_invsqrt_kernel(const float* __restrict__ cov, float* __restrict__ sinv) {
  __shared__ float Ym[64 * LS];
  __shared__ float Zm[64 * LS];
  __shared__ float Wm[64 * LS];
  __shared__ float red[512];
  __shared__ float sc[3];
  const int k = blockIdx.x, tid = threadIdx.x;
  const int lane = tid & 31, wave = tid >> 5;
  const int ti = wave >> 2, tj = wave & 3;
  const float* C = cov + (size_t)k * 4096;

  // async-stage C into Wm
#pragma unroll
  for (int tseg = 0; tseg < 2; ++tseg) {
    const int s = tid + tseg * 512;
    const int i = s >> 4, q = (s & 15) << 2;
    async_copy16(C + (size_t)i * 64 + q, &Wm[i * LS + q]);
  }
  wait_async0();
  __syncthreads();

  float ps = 0.f;
  for (int e = tid; e < 4096; e += 512) {
    const float v = Wm[(e >> 6) * LS + (e & 63)];
    ps += v * v;
  }
  red[tid] = ps;
  __syncthreads();
  for (int s = 256; s > 0; s >>= 1) {
    if (tid < s) red[tid] += red[tid + s];
    __syncthreads();
  }
  if (tid == 0) {
    float tr = 0.f;
    for (int i = 0; i < 64; ++i) tr += Wm[i * LS + i];
    const float frob2  = red[0];
    const float tm     = tr * (1.0f / 64.0f);     // mean(diag)
    const float second = frob2 * (1.0f / 64.0f);  // sum(C^2)/p
    const float num = tr * tr - second;
    const float den = (float)(BATCH - 1) * (second - tm * tm);
    sc[0] = fminf(num / den, 1.0f);               // rho
    sc[1] = tm;
    sc[2] = fmaxf(tr, 1e-30f);                    // trace (preserved by shrinkage)
  }
  __syncthreads();
  const float rho = sc[0], tm = sc[1], s = sc[2];
  const float invs = 1.0f / s;

  // Y0 = shrunk/trace (eigs in (0,1] -> NS converges), Z0 = I
  for (int e = tid; e < 4096; e += 512) {
    const int i = e >> 6, j = e & 63;
    const float v = (1.f - rho) * Wm[i * LS + j] + ((i == j) ? rho * tm : 0.f);
    Ym[i * LS + j] = v * invs;
    Zm[i * LS + j] = (i == j) ? 1.f : 0.f;
  }
  __syncthreads();

  for (int it = 0; it < NS_ITERS; ++it) {
    v8f t = lds_gemm64(Zm, Ym, ti, tj, lane);     // T = Z*Y
#pragma unroll
    for (int r = 0; r < 8; ++r) {
      const int row = ti * 16 + r + ((lane >> 4) << 3);
      const int col = tj * 16 + (lane & 15);
      Wm[row * LS + col] = ((row == col) ? 1.5f : 0.f) - 0.5f * t[r];
    }
    __syncthreads();
    v8f ny = lds_gemm64(Ym, Wm, ti, tj, lane);    // Y' = Y*W
    v8f nz = lds_gemm64(Wm, Zm, ti, tj, lane);    // Z' = W*Z
    __syncthreads();
#pragma unroll
    for (int r = 0; r < 8; ++r) {
      const int row = ti * 16 + r + ((lane >> 4) << 3);
      const int col = tj * 16 + (lane & 15);
      Ym[row * LS + col] = ny[r];
      Zm[row * LS + col] = nz[r];
    }
    __syncthreads();
  }

  const float rs = rsqrtf(s);                     // C^{-1/2} = Z / sqrt(trace)
  for (int e = tid; e < 4096; e += 512) {
    const int i = e >> 6, j = e & 63;
    sinv[(size_t)k * 4096 + e] = Zm[i * LS + j] * rs;
  }
}

// ---------------- stage 5: y[k,:,btile] = S_inv[k] * y[k,:,btile]  (in place)
__global__ void __launch_bounds__(512)
whiten_kernel(const float* __restrict__ sinv, float* __restrict__ y) {
  __shared__ float Sm[64 * LS];
  __shared__ float Yt[64 * LS];
  const int k  = blockIdx.y;
  const int b0 = blockIdx.x * 64;
  const int tid = threadIdx.x;
  const int lane = tid & 31, wave = tid >> 5;
  const int ti = wave >> 2, tj = wave & 3;
  float* yk = y + (size_t)k * (DDIM * BATCH);
  const float* S = sinv + (size_t)k * 4096;
#pragma unroll
  for (int tseg = 0; tseg < 2; ++tseg) {
    const int s = tid + tseg * 512;
    const int i = s >> 4, q = (s & 15) << 2;
    async_copy16(S + (size_t)i * 64 + q, &Sm[i * LS + q]);
    async_copy16(yk + (size_t)i * BATCH + b0 + q, &Yt[i * LS + q]);
  }
  wait_async0();
  __syncthreads();
  v8f acc = lds_gemm64(Sm, Yt, ti, tj, lane);
#pragma unroll
  for (int r = 0; r < 8; ++r) {
    const int row = ti * 16 + r + ((lane >> 4) << 3);
    const int col = tj * 16 + (lane & 15);
    yk[(size_t)row * BATCH + b0 + col] = acc[r];  // block wrote only what it staged
  }
}

// ---------------- stage 6: transpose back y[K,D,B] -> out[B,D,K]
__global__ void __launch_bounds__(256)
out_transpose_kernel(const float* __restrict__ y, float* __restrict__ out) {
  __shared__ float t[64 * LS];
  const int d = blockIdx.y, b0 = blockIdx.x * 64, tid = threadIdx.x;
#pragma unroll
  for (int tseg = 0; tseg < 4; ++tseg) {
    const int s = tid + tseg * 256;
    const int kk = s >> 4, q = (s & 15) << 2;
    async_copy16(y + (size_t)kk * (DDIM * BATCH) + (size_t)d * BATCH + b0 + q,
                 &t[kk * LS + q]);
  }
  wait_async0();
  __syncthreads();
  for (int e = tid; e < 64 * 64; e += 256) {
    const int i = e >> 6, j = e & 63;            // j (=k) fastest -> coalesced writes
    out[(size_t)(b0 + i) * (DDIM * KCL) + d * KCL + j] = t[j * LS + i];
  }
}

extern "C" void kernel_launch(void* const* d_in, const int* in_sizes, int n_in,
                              void* d_out, int out_size, void* d_ws, size_t ws_size,
                              hipStream_t stream) {
  (void)in_sizes; (void)n_in; (void)out_size; (void)ws_size;
  const float* x = (const float*)d_in[0];
  float* out = (float*)d_out;
  float* ws  = (float*)d_ws;

  // workspace layout (floats):
  float* partial = ws;                           // 32*4096        = 131072
  float* musum   = partial + 32 * 4096;          // 4096
  float* covp    = musum + 4096;                 // 16*64*4096     = 4194304
  float* cov     = covp + (size_t)BSPLIT * KCL * 4096;  // 262144
  float* sinv    = cov + (size_t)KCL * 4096;     // 262144
  float* y       = sinv + (size_t)KCL * 4096;    // 64*64*8192     = 33554432 (~154 MB total)

  mean_partial_kernel    <<<dim3(16, 32), dim3(256), 0, stream>>>(x, partial);
  mean_reduce_kernel     <<<dim3(16),     dim3(256), 0, stream>>>(partial, musum);
  center_transpose_kernel<<<dim3(BATCH / 64, DDIM), dim3(256), 0, stream>>>(x, musum, y);
  cov_partial_kernel     <<<dim3(KCL, BSPLIT), dim3(512), 0, stream>>>(y, covp);
  cov_reduce_kernel      <<<dim3(KCL, 16), dim3(256), 0, stream>>>(covp, cov);
  ns_invsqrt_kernel      <<<dim3(KCL),    dim3(512), 0, stream>>>(cov, sinv);
  whiten_kernel          <<<dim3(BATCH / 64, KCL), dim3(512), 0, stream>>>(sinv, y);
  out_transpose_kernel   <<<dim3(BATCH / 64, DDIM), dim3(256), 0, stream>>>(y, out);
}